// MSGCN_15375982920184
// MI455X (gfx1250) — compile-verified
//
#include <hip/hip_runtime.h>
#include <hip/hip_bf16.h>

#define N_USERS 50000
#define N_ITEMS 50000
#define N_NODES (N_USERS + N_ITEMS)
#define N_EDGES 1600000
#define N_PAIRS 200000
#define HID 64
#define UDIM 128

typedef __attribute__((ext_vector_type(2))) float v2f;
typedef __attribute__((ext_vector_type(8))) float v8f;

__device__ __forceinline__ v8f wmma_k4(v2f a, v2f b, v8f c) {
  // D = A(16x4 f32) * B(4x16 f32) + C(16x16 f32)
  return __builtin_amdgcn_wmma_f32_16x16x4_f32(false, a, false, b, (short)0, c,
                                               false, false);
}

// ---------------- degree / norm ----------------
__global__ void degree_kernel(const int* __restrict__ esrc,
                              const int* __restrict__ edst,
                              float* __restrict__ outdeg,
                              float* __restrict__ indeg) {
  int e = blockIdx.x * blockDim.x + threadIdx.x;
  if (e >= N_EDGES) return;
  atomicAdd(&outdeg[esrc[e]], 1.0f);
  atomicAdd(&indeg[edst[e]], 1.0f);
}

__global__ void norm_kernel(const float* __restrict__ outdeg,
                            const float* __restrict__ indeg,
                            float* __restrict__ onorm,
                            float* __restrict__ inorm) {
  int n = blockIdx.x * blockDim.x + threadIdx.x;
  if (n >= N_NODES) return;
  onorm[n] = rsqrtf(fmaxf(outdeg[n], 1.0f));
  inorm[n] = rsqrtf(fmaxf(indeg[n], 1.0f));
}

// ---------------- embed GEMM: out = relu(X[M,128] @ W[128,64] + b) ----------------
// one wave per 16-row tile, computes 16x64 via 4 accumulators, K=128 in k=4 steps
__global__ void embed_kernel(const float* __restrict__ X,
                             const float* __restrict__ W,
                             const float* __restrict__ bvec,
                             float* __restrict__ out, int M) {
  int wave = (blockIdx.x * blockDim.x + threadIdx.x) >> 5;
  int lane = threadIdx.x & 31;
  int row0 = wave * 16;
  if (row0 >= M) return;                       // wave-uniform exit
  int m = lane & 15;
  int kb = 2 * (lane >> 4);
  const float* arow = X + (size_t)(row0 + m) * UDIM;

  v8f acc0 = {}, acc1 = {}, acc2 = {}, acc3 = {};
  for (int k = 0; k < UDIM; k += 4) {
    int kk = k + kb;
    v2f a;
    a.x = arow[kk];
    a.y = arow[kk + 1];
    const float* w0 = W + (size_t)kk * HID;
    const float* w1 = w0 + HID;
    v2f b;
    b.x = w0[0 * 16 + m]; b.y = w1[0 * 16 + m]; acc0 = wmma_k4(a, b, acc0);
    b.x = w0[1 * 16 + m]; b.y = w1[1 * 16 + m]; acc1 = wmma_k4(a, b, acc1);
    b.x = w0[2 * 16 + m]; b.y = w1[2 * 16 + m]; acc2 = wmma_k4(a, b, acc2);
    b.x = w0[3 * 16 + m]; b.y = w1[3 * 16 + m]; acc3 = wmma_k4(a, b, acc3);
  }
  int nb = lane & 15;
  int mtop = 8 * (lane >> 4);
  v8f accs[4] = {acc0, acc1, acc2, acc3};
#pragma unroll
  for (int t = 0; t < 4; ++t) {
    int n = t * 16 + nb;
    float bn = bvec[n];
#pragma unroll
    for (int r = 0; r < 8; ++r) {
      float v = accs[t][r] + bn;
      out[(size_t)(row0 + mtop + r) * HID + n] = fmaxf(v, 0.0f);
    }
  }
}

// ---------------- GCN scatter: agg[dst] += emb[src]*out_norm[src] ----------------
__global__ void scatter_kernel(const float* __restrict__ emb,
                               const float* __restrict__ onorm,
                               const int* __restrict__ esrc,
                               const int* __restrict__ edst,
                               float* __restrict__ agg) {
  long long idx = (long long)blockIdx.x * blockDim.x + threadIdx.x;
  int e = (int)(idx >> 6);
  int h = (int)(idx & 63);
  if (e >= N_EDGES) return;
  int s = esrc[e];
  int d = edst[e];
  float v = emb[(size_t)s * HID + h] * onorm[s];
  atomicAdd(&agg[(size_t)d * HID + h], v);
}

// ---------------- post: e = agg*in_norm; layer_out = e; emb_acc += e*scale ----------------
__global__ void post_kernel(const float* __restrict__ agg,
                            const float* __restrict__ inorm,
                            float* __restrict__ layer_out,
                            float* __restrict__ emb_acc, float scale) {
  long long idx = (long long)blockIdx.x * blockDim.x + threadIdx.x;
  if (idx >= (long long)N_NODES * HID) return;
  int n = (int)(idx >> 6);
  float e = agg[idx] * inorm[n];
  layer_out[idx] = e;
  emb_acc[idx] += e * scale;
}

__global__ void add_side_kernel(float* __restrict__ iemb,
                                const float* __restrict__ side) {
  long long idx = (long long)blockIdx.x * blockDim.x + threadIdx.x;
  if (idx >= (long long)N_ITEMS * HID) return;
  iemb[idx] += side[idx];
}

// ---------------- decode: logits = elu([uE|iE|T] @ W1 + b1) @ W2 ----------------
// one wave per 16-pair tile; K=128 WMMA over [uE|iE], T*W1[128,:] rank-1 in epilogue
__global__ void decode_kernel(const float* __restrict__ uemb,
                              const float* __restrict__ iemb,
                              const int* __restrict__ userId,
                              const int* __restrict__ posId,
                              const int* __restrict__ negId,
                              const float* __restrict__ T,
                              const float* __restrict__ W1,
                              const float* __restrict__ b1,
                              const float* __restrict__ W2,
                              float* __restrict__ logits) {
  int wave = (blockIdx.x * blockDim.x + threadIdx.x) >> 5;
  int lane = threadIdx.x & 31;
  int row0 = wave * 16;
  if (row0 >= 2 * N_PAIRS) return;             // wave-uniform exit
  int m = lane & 15;
  int kb = 2 * (lane >> 4);
  int r = row0 + m;
  int p = (r < N_PAIRS) ? r : r - N_PAIRS;
  int uid = userId[p];
  int iid = (r < N_PAIRS) ? posId[p] : negId[r - N_PAIRS];
  const float* ua = uemb + (size_t)uid * HID;
  const float* ia = iemb + (size_t)iid * HID;

  v8f acc0 = {}, acc1 = {}, acc2 = {}, acc3 = {};
  for (int k = 0; k < 2 * HID; k += 4) {
    int kk = k + kb;                            // even; kk,kk+1 on same half
    const float* src = (kk < HID) ? (ua + kk) : (ia + kk - HID);
    v2f a;
    a.x = src[0];
    a.y = src[1];
    const float* w0 = W1 + (size_t)kk * HID;
    const float* w1 = w0 + HID;
    v2f b;
    b.x = w0[0 * 16 + m]; b.y = w1[0 * 16 + m]; acc0 = wmma_k4(a, b, acc0);
    b.x = w0[1 * 16 + m]; b.y = w1[1 * 16 + m]; acc1 = wmma_k4(a, b, acc1);
    b.x = w0[2 * 16 + m]; b.y = w1[2 * 16 + m]; acc2 = wmma_k4(a, b, acc2);
    b.x = w0[3 * 16 + m]; b.y = w1[3 * 16 + m]; acc3 = wmma_k4(a, b, acc3);
  }

  int nb = lane & 15;
  int mtop = 8 * (lane >> 4);
  float Tm[8];
#pragma unroll
  for (int rr = 0; rr < 8; ++rr) Tm[rr] = T[row0 + mtop + rr];

  v8f accs[4] = {acc0, acc1, acc2, acc3};
  float total[8] = {0, 0, 0, 0, 0, 0, 0, 0};
#pragma unroll
  for (int t = 0; t < 4; ++t) {
    int n = t * 16 + nb;
    float w1T = W1[(size_t)(2 * HID) * HID + n]; // row 128 (T column weight)
    float b1n = b1[n];
    float w2n = W2[n];
#pragma unroll
    for (int rr = 0; rr < 8; ++rr) {
      float h = accs[t][rr] + Tm[rr] * w1T + b1n;
      float e = (h > 0.0f) ? h : (__expf(h) - 1.0f);   // elu
      total[rr] += e * w2n;
    }
  }
#pragma unroll
  for (int rr = 0; rr < 8; ++rr) {
    float s = total[rr];
    s += __shfl_xor(s, 8, 16);
    s += __shfl_xor(s, 4, 16);
    s += __shfl_xor(s, 2, 16);
    s += __shfl_xor(s, 1, 16);
    if (nb == 0) logits[row0 + mtop + rr] = s;
  }
}

extern "C" void kernel_launch(void* const* d_in, const int* in_sizes, int n_in,
                              void* d_out, int out_size, void* d_ws,
                              size_t ws_size, hipStream_t stream) {
  const float* user_features = (const float*)d_in[0];
  const float* item_features = (const float*)d_in[1];
  const float* item_side     = (const float*)d_in[2];
  const float* T_f  = (const float*)d_in[3];
  const float* T_cf = (const float*)d_in[4];
  const float* Wu = (const float*)d_in[5];
  const float* bu = (const float*)d_in[6];
  const float* Wi = (const float*)d_in[7];
  const float* bi = (const float*)d_in[8];
  const float* W1 = (const float*)d_in[9];
  const float* b1 = (const float*)d_in[10];
  const float* W2 = (const float*)d_in[11];
  const int* esrc = (const int*)d_in[12];
  const int* edst = (const int*)d_in[13];
  const int* userId = (const int*)d_in[14];
  const int* posId  = (const int*)d_in[15];
  const int* negId  = (const int*)d_in[16];

  // d_out: [user_emb | item_emb | logits_f | logits_cf]
  float* uemb = (float*)d_out;
  float* iemb = uemb + (size_t)N_USERS * HID;
  float* logits_f  = iemb + (size_t)N_ITEMS * HID;
  float* logits_cf = logits_f + 2 * N_PAIRS;
  float* emb_all = uemb;  // user||item contiguous == concatenate

  // workspace
  float* agg    = (float*)d_ws;                       // N_NODES*HID
  float* wse    = agg + (size_t)N_NODES * HID;        // N_NODES*HID (layer output)
  float* outdeg = wse + (size_t)N_NODES * HID;        // N_NODES
  float* indeg  = outdeg + N_NODES;
  float* onorm  = indeg + N_NODES;
  float* inorm  = onorm + N_NODES;

  // degrees + norms
  hipMemsetAsync(outdeg, 0, (size_t)2 * N_NODES * sizeof(float), stream);
  degree_kernel<<<(N_EDGES + 255) / 256, 256, 0, stream>>>(esrc, edst, outdeg, indeg);
  norm_kernel<<<(N_NODES + 255) / 256, 256, 0, stream>>>(outdeg, indeg, onorm, inorm);

  // initial embeddings (WMMA GEMM + relu), written straight into d_out
  {
    int tiles = N_USERS / 16;                          // 3125
    int blocks = (tiles + 7) / 8;                      // 8 waves/block
    embed_kernel<<<blocks, 256, 0, stream>>>(user_features, Wu, bu, uemb, N_USERS);
    embed_kernel<<<blocks, 256, 0, stream>>>(item_features, Wi, bi, iemb, N_ITEMS);
  }

  // two GCN layers
  long long nelem = (long long)N_NODES * HID;
  long long edgework = (long long)N_EDGES * HID;
  for (int layer = 0; layer < 2; ++layer) {
    hipMemsetAsync(agg, 0, (size_t)nelem * sizeof(float), stream);
    const float* src_emb = (layer == 0) ? emb_all : wse;
    scatter_kernel<<<(int)((edgework + 255) / 256), 256, 0, stream>>>(
        src_emb, onorm, esrc, edst, agg);
    float scale = 1.0f / (float)(layer + 2);
    post_kernel<<<(int)((nelem + 255) / 256), 256, 0, stream>>>(
        agg, inorm, wse, emb_all, scale);
  }

  // item side features
  add_side_kernel<<<(int)(((long long)N_ITEMS * HID + 255) / 256), 256, 0, stream>>>(
      iemb, item_side);

  // decode heads (WMMA GEMM + elu + in-wave W2 reduction)
  {
    int tiles = (2 * N_PAIRS) / 16;                    // 25000
    int blocks = (tiles + 7) / 8;
    decode_kernel<<<blocks, 256, 0, stream>>>(uemb, iemb, userId, posId, negId,
                                              T_f, W1, b1, W2, logits_f);
    decode_kernel<<<blocks, 256, 0, stream>>>(uemb, iemb, userId, posId, negId,
                                              T_cf, W1, b1, W2, logits_cf);
  }
}